// gnn_binary_classifier_28896539968012
// MI455X (gfx1250) — compile-verified
//
#include <hip/hip_runtime.h>
#include <math.h>

// ---------------------------------------------------------------------------
// GNN binary classifier on MI455X (gfx1250, wave32).
//
// Roofline: ~1.7 GFLOP vs ~234 MB unavoidable HBM traffic (GRU weights 100MB
// read once -> L2-resident across the 8 steps; fc1_w 134MB read once).
// => pure bandwidth problem (~10us at 23.3 TB/s). We use fp32 WMMA
// (V_WMMA_F32_16X16X4_F32) with M padded 4->16; rows 0-3 of the A fragment
// carry a=A@h and rows 4-7 carry h, so ONE pass over W_ih and W_hh per step
// produces both gi (C1 rows 0-3) and gh (C2 rows 4-7).
// ---------------------------------------------------------------------------

typedef __attribute__((ext_vector_type(2))) float v2f;
typedef __attribute__((ext_vector_type(8))) float v8f;

#define DD    2048          // state dim
#define HH    4096          // hidden dim
#define NNODE 4
#define TD    (3 * DD)      // 6144 gate width
#define STEPS 8

// workspace layout (float offsets); total 94208 floats = 368 KB
#define X_OFF   0                   // X[16][DD]: rows0-3=a, rows4-7=h, rows8-15=0
#define H_OFF   (16 * DD)           // h[4][DD]
#define GI_OFF  (H_OFF + 4 * DD)    // gi[4][TD]
#define GH_OFF  (GI_OFF + 4 * TD)   // gh[4][TD]
#define X1_OFF  (GH_OFF + 4 * TD)   // x1[HH]

// ---------------------------------------------------------------------------
// Step 0: h = annotation; build X (a = A@annotation in rows 0-3, h in 4-7,
// zeros in 8-15). Runs every launch -> no reliance on prior workspace state.
// ---------------------------------------------------------------------------
__global__ __launch_bounds__(256) void init_kernel(const float* __restrict__ Amat,
                                                   const float* __restrict__ ann,
                                                   float* __restrict__ ws) {
  int k = blockIdx.x * blockDim.x + threadIdx.x;
  if (k >= DD) return;
  float* X    = ws + X_OFF;
  float* hbuf = ws + H_OFF;
  float h0 = ann[0 * DD + k], h1 = ann[1 * DD + k];
  float h2 = ann[2 * DD + k], h3 = ann[3 * DD + k];
  hbuf[0 * DD + k] = h0; hbuf[1 * DD + k] = h1;
  hbuf[2 * DD + k] = h2; hbuf[3 * DD + k] = h3;
#pragma unroll
  for (int m = 0; m < 4; ++m) {
    float a = Amat[m * 4 + 0] * h0 + Amat[m * 4 + 1] * h1 +
              Amat[m * 4 + 2] * h2 + Amat[m * 4 + 3] * h3;
    X[m * DD + k] = a;
  }
  X[(4 + 0) * DD + k] = h0; X[(4 + 1) * DD + k] = h1;
  X[(4 + 2) * DD + k] = h2; X[(4 + 3) * DD + k] = h3;
#pragma unroll
  for (int m = 8; m < 16; ++m) X[m * DD + k] = 0.0f;
}

// ---------------------------------------------------------------------------
// Per step: gi[4][TD] = a @ W_ih^T, gh[4][TD] = h @ W_hh^T via fp32 WMMA.
// Block = 4 waves; each block owns a 16-column output tile (384 blocks).
// K=2048 split 512-per-wave; cross-wave reduce through 2 KB LDS.
// Fragment layouts (ISA 7.12.2, f32 16x16x4):
//   A: lane = (M=lane&15, K=k0+2*(lane>>4)+{0,1})  -> float2 from X row M
//   B: lane = (N=lane&15, K=k0+2*(lane>>4)+{0,1})  -> float2 from W row n0+N
// ---------------------------------------------------------------------------
__global__ __launch_bounds__(128) void gru_gemv_kernel(const float* __restrict__ Wih,
                                                       const float* __restrict__ Whh,
                                                       float* __restrict__ ws) {
  const float* X  = ws + X_OFF;
  float* gi = ws + GI_OFF;
  float* gh = ws + GH_OFF;

  const int n0   = blockIdx.x * 16;
  const int tid  = threadIdx.x;
  const int wv   = tid >> 5;        // wave in block (0..3)
  const int lane = tid & 31;
  const int l16  = lane & 15;
  const int half = lane >> 4;

  const int kbeg = wv * (DD / 4);   // 512-wide K slice per wave
  const int koff = kbeg + 2 * half;

  const float* xrow = X + l16 * DD;
  const float* wihr = Wih + (size_t)(n0 + l16) * DD;
  const float* whhr = Whh + (size_t)(n0 + l16) * DD;

  v8f c1 = {};  // rows 0-3 = gi partial
  v8f c2 = {};  // rows 4-7 = gh partial

#pragma unroll 4
  for (int it = 0; it < DD / 16; ++it) {          // 128 K-steps of 4
    const int kk = koff + it * 4;
    v2f aa = *(const v2f*)(xrow + kk);
    v2f bi = *(const v2f*)(wihr + kk);
    v2f bh = *(const v2f*)(whhr + kk);
    c1 = __builtin_amdgcn_wmma_f32_16x16x4_f32(false, aa, false, bi,
                                               (short)0, c1, false, false);
    c2 = __builtin_amdgcn_wmma_f32_16x16x4_f32(false, aa, false, bh,
                                               (short)0, c2, false, false);
  }

  __shared__ float red[4][8][16];
  if (lane < 16) {
#pragma unroll
    for (int m = 0; m < 4; ++m) {
      red[wv][m][l16]     = c1[m];      // gi row m, col n0+l16
      red[wv][4 + m][l16] = c2[4 + m];  // gh row m, col n0+l16
    }
  }
  __syncthreads();

  const int r8  = tid >> 4;   // 0..7
  const int col = tid & 15;
  float s = red[0][r8][col] + red[1][r8][col] + red[2][r8][col] + red[3][r8][col];
  if (r8 < 4) gi[r8 * TD + n0 + col] = s;
  else        gh[(r8 - 4) * TD + n0 + col] = s;
}

// ---------------------------------------------------------------------------
// GRU gates + message passing for the next step. One thread per column k.
// ---------------------------------------------------------------------------
__global__ __launch_bounds__(256) void gru_gate_kernel(const float* __restrict__ Amat,
                                                       float* __restrict__ ws) {
  int k = blockIdx.x * blockDim.x + threadIdx.x;
  if (k >= DD) return;
  float* X    = ws + X_OFF;
  float* hbuf = ws + H_OFF;
  const float* gi = ws + GI_OFF;
  const float* gh = ws + GH_OFF;

  float hn[4];
#pragma unroll
  for (int nd = 0; nd < 4; ++nd) {
    float ir = gi[nd * TD + k];
    float iz = gi[nd * TD + DD + k];
    float in_ = gi[nd * TD + 2 * DD + k];
    float hr = gh[nd * TD + k];
    float hz = gh[nd * TD + DD + k];
    float hnn = gh[nd * TD + 2 * DD + k];
    float r = 1.0f / (1.0f + expf(-(ir + hr)));
    float z = 1.0f / (1.0f + expf(-(iz + hz)));
    float n = tanhf(in_ + r * hnn);
    float hold = hbuf[nd * DD + k];
    hn[nd] = (1.0f - z) * n + z * hold;
  }
#pragma unroll
  for (int nd = 0; nd < 4; ++nd) hbuf[nd * DD + k] = hn[nd];
#pragma unroll
  for (int m = 0; m < 4; ++m) {
    float a = Amat[m * 4 + 0] * hn[0] + Amat[m * 4 + 1] * hn[1] +
              Amat[m * 4 + 2] * hn[2] + Amat[m * 4 + 3] * hn[3];
    X[m * DD + k]       = a;      // next-step a
    X[(4 + m) * DD + k] = hn[m];  // next-step h
  }
  // rows 8-15 stay zero (written by init_kernel this launch)
}

// ---------------------------------------------------------------------------
// fc1: x1[j] = dot(h.flatten(), fc1_w[j]) + fc1_b[j].  M=1 GEMV: one wave
// per row, float4-coalesced stream (134 MB -> the big HBM read).
// ---------------------------------------------------------------------------
__global__ __launch_bounds__(256) void fc1_kernel(const float* __restrict__ f1w,
                                                  const float* __restrict__ f1b,
                                                  float* __restrict__ ws) {
  const float* enc = ws + H_OFF;   // [4][DD] contiguous == reshape(1,-1)
  float* x1 = ws + X1_OFF;
  const int wv   = threadIdx.x >> 5;
  const int lane = threadIdx.x & 31;
  const int j = blockIdx.x * 8 + wv;   // 512 blocks * 8 waves = 4096 rows
  const float* row = f1w + (size_t)j * (NNODE * DD);

  float acc = 0.0f;
  for (int i = lane; i < (NNODE * DD) / 4; i += 32) {
    float4 w4 = *(const float4*)(row + i * 4);
    float4 e4 = *(const float4*)(enc + i * 4);
    acc += w4.x * e4.x + w4.y * e4.y + w4.z * e4.z + w4.w * e4.w;
  }
#pragma unroll
  for (int off = 16; off > 0; off >>= 1) acc += __shfl_down(acc, off, 32);
  if (lane == 0) x1[j] = acc + f1b[j];
}

// ---------------------------------------------------------------------------
// fc2 + log_softmax, single block.
// ---------------------------------------------------------------------------
__global__ __launch_bounds__(256) void fc2_kernel(const float* __restrict__ f2w,
                                                  const float* __restrict__ f2b,
                                                  const float* __restrict__ ws,
                                                  float* __restrict__ out) {
  __shared__ float s0a[256], s1a[256];
  const float* x1 = ws + X1_OFF;
  const int t = threadIdx.x;
  float s0 = 0.0f, s1 = 0.0f;
  for (int i = t; i < HH; i += 256) {
    float v = x1[i];
    s0 += v * f2w[i];
    s1 += v * f2w[HH + i];
  }
  s0a[t] = s0; s1a[t] = s1;
  __syncthreads();
  for (int st = 128; st > 0; st >>= 1) {
    if (t < st) { s0a[t] += s0a[t + st]; s1a[t] += s1a[t + st]; }
    __syncthreads();
  }
  if (t == 0) {
    float a0 = s0a[0] + f2b[0];
    float a1 = s1a[0] + f2b[1];
    float m = fmaxf(a0, a1);
    float lse = m + logf(expf(a0 - m) + expf(a1 - m));
    out[0] = a0 - lse;
    out[1] = a1 - lse;
  }
}

// ---------------------------------------------------------------------------
extern "C" void kernel_launch(void* const* d_in, const int* in_sizes, int n_in,
                              void* d_out, int out_size, void* d_ws, size_t ws_size,
                              hipStream_t stream) {
  const float* Amat = (const float*)d_in[0];
  const float* ann  = (const float*)d_in[1];
  const float* Wih  = (const float*)d_in[2];
  const float* Whh  = (const float*)d_in[3];
  const float* f1w  = (const float*)d_in[4];
  const float* f1b  = (const float*)d_in[5];
  const float* f2w  = (const float*)d_in[6];
  const float* f2b  = (const float*)d_in[7];
  // d_in[8] = GRU_step (== 8 in the reference); fixed at compile time.
  float* ws  = (float*)d_ws;       // needs >= 376,832 bytes
  float* out = (float*)d_out;

  init_kernel<<<DD / 256, 256, 0, stream>>>(Amat, ann, ws);
  for (int s = 0; s < STEPS; ++s) {
    gru_gemv_kernel<<<TD / 16, 128, 0, stream>>>(Wih, Whh, ws);
    gru_gate_kernel<<<DD / 256, 256, 0, stream>>>(Amat, ws);
  }
  fc1_kernel<<<HH / 8, 256, 0, stream>>>(f1w, f1b, ws);
  fc2_kernel<<<1, 256, 0, stream>>>(f2w, f2b, ws, out);
}